// PriSTI_38551626449548
// MI455X (gfx1250) — compile-verified
//
#include <hip/hip_runtime.h>
#include <math.h>

// PriSTI forward for MI455X (gfx1250, wave32). All GEMMs via V_WMMA_F32_16X16X4_F32
// (fp32 matrix pipe: model is L2-resident / bandwidth bound, fp32 keeps reference
// numerics). Deterministic: no atomics (CSR-based graph conv, block reductions).
// Workspace need: ~204 MB.
//
// GEMM kernels:
//  - k_lin<TN>: row-major X[M,K] @ W[N,K]^T, M%128==0, K%4==0, N=TN*16.
//    256-thread block (8 waves). W is staged ONCE per block into LDS with
//    GLOBAL_LOAD_ASYNC_TO_LDS_B128 (ASYNCcnt + s_wait_asynccnt + barrier) —
//    8x reduction of W traffic. Each wave owns a 16-row stripe: A fragments
//    via global_load_b64 (+ global_prefetch ahead), B fragments via
//    ds_load_b64, TN independent WMMA accumulators pipeline back-to-back.
//  - k_gemm: generic strided/batched with CLAMPED (branch-free) edge handling
//    for the K=207 encoder and M=207 decoder GEMMs.

namespace {

constexpr int       kB    = 4;
constexpr int       kT    = 64;
constexpr int       kN    = 207;
constexpr int       kH    = 8;
constexpr int       kE    = 3312;
constexpr int       kRows = kB * kT * kN;              // 52992
constexpr long long kPerB = (long long)kT * kN * 64;   // 847872 elems per batch
constexpr long long kGsz  = kPerB / 4;                 // 211968 per group
constexpr int       kTN   = kT * kN;                   // 13248
constexpr float     kEps  = 1e-5f;
constexpr size_t    kSlot = (size_t)kRows * 64;        // 3,391,488 floats

typedef __attribute__((ext_vector_type(2))) float v2f;
typedef __attribute__((ext_vector_type(8))) float v8f;

__device__ __forceinline__ float geluf(float x) {
  return 0.5f * x * (1.f + erff(x * 0.7071067811865475f));
}

// ---------------------------------------------------------------------------
// Fast row-major linear: C[M,N] = act(alpha * A[M,K] @ W[N,K]^T + bias) (+=).
// Requirements: M%128==0, K%4==0, N==TN*16, N*K<=8192.
// Block: 256 threads = 8 waves; wave w handles rows [blk*128+w*16, +16).
// ---------------------------------------------------------------------------
template <int TN>
__global__ __launch_bounds__(256) void k_lin(const float* __restrict__ A, int K,
                                             const float* __restrict__ W,
                                             const float* __restrict__ bias,
                                             float* __restrict__ Cm,
                                             int act, int acc, float alpha)
{
  const int N = TN * 16;
  __shared__ __align__(16) float sW[8192];

  // ---- async-stage W[N*K] into LDS (16B granules, ASYNCcnt-tracked) ----
  {
    const int nvec = (N * K) >> 2;                       // # of 16-byte chunks
    unsigned ldsBase = (unsigned)(unsigned long long)(void*)sW;  // low 32b = LDS offset
    for (int i = threadIdx.x; i < nvec; i += 256) {
      unsigned lo = ldsBase + (unsigned)(i * 16);
      unsigned long long ga = (unsigned long long)(W + (size_t)i * 4);
      asm volatile("global_load_async_to_lds_b128 %0, %1, off"
                   :: "v"(lo), "v"(ga) : "memory");
    }
    asm volatile("s_wait_asynccnt 0" ::: "memory");
  }
  __syncthreads();

  const int wv   = threadIdx.x >> 5;
  const int lane = threadIdx.x & 31;
  const int half = lane >> 4;       // 0: k pair {0,1}, 1: k pair {2,3}
  const int l16  = lane & 15;
  const int tm   = blockIdx.x * 128 + wv * 16;

  v8f cf[TN];
#pragma unroll
  for (int j = 0; j < TN; ++j) cf[j] = (v8f){};

  const float* Arow = A + (size_t)(tm + l16) * K + half * 2;

  for (int k0 = 0; k0 < K; k0 += 4) {
    __builtin_prefetch(Arow + k0 + 32, 0, 0);            // global_prefetch ahead
    float2 a2 = *(const float2*)(Arow + k0);
    v2f av = {a2.x, a2.y};
#pragma unroll
    for (int j = 0; j < TN; ++j) {
      float2 b2 = *(const float2*)(&sW[(j * 16 + l16) * K + half * 2 + k0]);
      v2f bv = {b2.x, b2.y};
      cf[j] = __builtin_amdgcn_wmma_f32_16x16x4_f32(false, av, false, bv,
                                                    (short)0, cf[j], false, false);
    }
  }

#pragma unroll
  for (int j = 0; j < TN; ++j) {
    int col = j * 16 + l16;
    float bc = bias ? bias[col] : 0.f;
#pragma unroll
    for (int r = 0; r < 8; ++r) {
      int row = tm + half * 8 + r;   // C frag: VGPR r -> M=r (lanes 0-15), r+8 (16-31)
      float v = cf[j][r] * alpha + bc;
      size_t off = (size_t)row * N + col;
      if (acc) v += Cm[off];
      if (act == 1)      v = fmaxf(v, 0.f);
      else if (act == 2) v = geluf(v);
      Cm[off] = v;
    }
  }
}

// ---------------------------------------------------------------------------
// Generic strided batched GEMM: C[m,n] = alpha * sum_k A[m,k]*B[k,n] (+bias)(+C)
// One wave32 per 16x16 C tile. Edge handling via clamped addresses + 0/1 mask
// (branch-free loads). biasMode: 0 none, 1 per-col, 2 per-row. act: 1 relu, 2 gelu.
// ---------------------------------------------------------------------------
__global__ void k_gemm(const float* __restrict__ A, long long aR, long long aC, long long aB,
                       const float* __restrict__ Bm, long long bR, long long bC, long long bB,
                       const float* __restrict__ bias, int biasMode,
                       float* __restrict__ Cm, long long cR, long long cC, long long cB,
                       int M, int N, int K, int tilesN, int act, int acc, float alpha)
{
  int tile = blockIdx.x;
  int tm = (tile / tilesN) * 16;
  int tn = (tile % tilesN) * 16;
  long long bz = blockIdx.y;
  A  += bz * aB;
  Bm += bz * bB;
  Cm += bz * cB;

  int lane = threadIdx.x & 31;
  int half = lane >> 4;
  int l16  = lane & 15;

  v8f cf = {};
  int ar  = tm + l16;
  int bn  = tn + l16;
  float rm = (ar < M) ? 1.f : 0.f;
  float cm = (bn < N) ? 1.f : 0.f;
  int arc = ar < M ? ar : M - 1;      // clamped (always-valid) addresses
  int bnc = bn < N ? bn : N - 1;
  const float* Abase = A + (long long)arc * aR;
  const float* Bbase = Bm + (long long)bnc * bC;

  for (int k0 = 0; k0 < K; k0 += 4) {
    int ka  = k0 + half * 2;
    int ka0 = ka < K ? ka : K - 1;
    int ka1 = ka + 1 < K ? ka + 1 : K - 1;
    float m0 = (ka     < K) ? 1.f : 0.f;
    float m1 = (ka + 1 < K) ? 1.f : 0.f;
    float a0 = Abase[(long long)ka0 * aC] * (rm * m0);
    float a1 = Abase[(long long)ka1 * aC] * (rm * m1);
    float b0 = Bbase[(long long)ka0 * bR] * (cm * m0);
    float b1 = Bbase[(long long)ka1 * bR] * (cm * m1);
    v2f av = {a0, a1};
    v2f bv = {b0, b1};
    cf = __builtin_amdgcn_wmma_f32_16x16x4_f32(false, av, false, bv, (short)0, cf,
                                               false, false);
  }

#pragma unroll
  for (int r = 0; r < 8; ++r) {
    int row = tm + half * 8 + r;
    int col = tn + l16;
    if (row < M && col < N) {
      float v = cf[r] * alpha;
      if (biasMode == 1)      v += bias[col];
      else if (biasMode == 2) v += bias[row];
      long long off = (long long)row * cR + (long long)col * cC;
      if (acc) v += Cm[off];
      if (act == 1)      v = fmaxf(v, 0.f);
      else if (act == 2) v = geluf(v);
      Cm[off] = v;
    }
  }
}

// ---------------------------------------------------------------------------
// Attention core: seq = 64, head dim = 8. One thread per query position.
// Instance g: base = (g/D2)*s1 + (g%D2)*s2 + head*8 ; seq element s at +s*ss.
// ---------------------------------------------------------------------------
__global__ void k_attn(const float* __restrict__ Q, const float* __restrict__ K,
                       const float* __restrict__ V, float* __restrict__ O,
                       int D2, long long s1, long long s2, long long ss)
{
  int g = blockIdx.x;
  int h = blockIdx.y;
  long long base = (long long)(g / D2) * s1 + (long long)(g % D2) * s2 + h * 8;
  int qi = threadIdx.x;   // 0..63

  const float* qp = Q + base + (long long)qi * ss;
  float q[8];
#pragma unroll
  for (int f = 0; f < 8; ++f) q[f] = qp[f];

  float sc[64];
  float mx = -1e30f;
  for (int s = 0; s < 64; ++s) {
    const float* kp = K + base + (long long)s * ss;
    float d = 0.f;
#pragma unroll
    for (int f = 0; f < 8; ++f) d += q[f] * kp[f];
    d *= 0.3535533905932738f;  // 1/sqrt(8)
    sc[s] = d;
    mx = fmaxf(mx, d);
  }
  float sum = 0.f;
  for (int s = 0; s < 64; ++s) { float e = expf(sc[s] - mx); sc[s] = e; sum += e; }
  float inv = 1.f / sum;
  float o[8] = {0.f, 0.f, 0.f, 0.f, 0.f, 0.f, 0.f, 0.f};
  for (int s = 0; s < 64; ++s) {
    float w = sc[s] * inv;
    const float* vp = V + base + (long long)s * ss;
#pragma unroll
    for (int f = 0; f < 8; ++f) o[f] += w * vp[f];
  }
  float* op = O + base + (long long)qi * ss;
#pragma unroll
  for (int f = 0; f < 8; ++f) op[f] = o[f];
}

// ---------------------------------------------------------------------------
// LayerNorm over last dim (64). One wave per row; optional second input added.
// ---------------------------------------------------------------------------
__global__ void k_ln(const float* __restrict__ a, const float* __restrict__ b,
                     const float* __restrict__ g, const float* __restrict__ be,
                     float* __restrict__ y, int rows)
{
  int lane = threadIdx.x & 31;
  long long row = (long long)blockIdx.x * 8 + (threadIdx.x >> 5);
  if (row >= rows) return;
  long long base = row * 64;
  int i0 = lane * 2, i1 = lane * 2 + 1;
  float x0 = a[base + i0], x1 = a[base + i1];
  if (b) { x0 += b[base + i0]; x1 += b[base + i1]; }
  float s = x0 + x1, q = x0 * x0 + x1 * x1;
  for (int o = 16; o > 0; o >>= 1) { s += __shfl_xor(s, o, 32); q += __shfl_xor(q, o, 32); }
  float mu  = s * (1.f / 64.f);
  float var = q * (1.f / 64.f) - mu * mu;
  float r   = rsqrtf(var + kEps);
  y[base + i0] = (x0 - mu) * r * g[i0] + be[i0];
  y[base + i1] = (x1 - mu) * r * g[i1] + be[i1];
}

// ---------------------------------------------------------------------------
// GroupNorm (faithful to reference: groups over flat [B,4,T*N*C/4], gamma by l/(T*N))
// ---------------------------------------------------------------------------
__global__ void k_gnstats(const float* __restrict__ a, const float* __restrict__ b,
                          float* __restrict__ stats)
{
  __shared__ float ss[256];
  __shared__ float sq[256];
  int gid = blockIdx.x;           // 0..15 = b*4+grp
  int bi = gid >> 2, gr = gid & 3;
  long long base = (long long)bi * kPerB + (long long)gr * kGsz;
  float s = 0.f, q = 0.f;
  for (long long i = threadIdx.x; i < kGsz; i += 256) {
    float x = a[base + i];
    if (b) x += b[base + i];
    s += x; q += x * x;
  }
  ss[threadIdx.x] = s; sq[threadIdx.x] = q;
  __syncthreads();
  for (int o = 128; o > 0; o >>= 1) {
    if (threadIdx.x < o) { ss[threadIdx.x] += ss[threadIdx.x + o]; sq[threadIdx.x] += sq[threadIdx.x + o]; }
    __syncthreads();
  }
  if (threadIdx.x == 0) {
    float mu  = ss[0] / (float)kGsz;
    float var = sq[0] / (float)kGsz - mu * mu;
    stats[gid * 2] = mu; stats[gid * 2 + 1] = var;
  }
}

__global__ void k_gnapply(const float* __restrict__ a, const float* __restrict__ b,
                          const float* __restrict__ stats, const float* __restrict__ g,
                          const float* __restrict__ be, float* __restrict__ y,
                          int relu, int total)
{
  int i = blockIdx.x * blockDim.x + threadIdx.x;
  if (i >= total) return;
  int bi  = i / (int)kPerB;
  int l   = i - bi * (int)kPerB;
  int gid = bi * 4 + (int)(l / (int)kGsz);
  int ci  = l / kTN;
  float mu = stats[gid * 2], var = stats[gid * 2 + 1];
  float x = a[i];
  if (b) x += b[i];
  float v = (x - mu) * rsqrtf(var + kEps) * g[ci] + be[ci];
  if (relu) v = fmaxf(v, 0.f);
  y[i] = v;
}

// ---------------------------------------------------------------------------
// Graph conv helpers (deterministic CSR build, single block)
// ---------------------------------------------------------------------------
__global__ void k_csr(const int* __restrict__ edges, const float* __restrict__ ew,
                      float* __restrict__ deg, float* __restrict__ nw,
                      int* __restrict__ cnt, int* __restrict__ off, int* __restrict__ cidx)
{
  const int* dst = edges + kE;
  int tid = threadIdx.x;
  for (int n = tid; n < kN; n += blockDim.x) {
    int cc = 0; float d = 0.f;
    for (int e = 0; e < kE; ++e) if (dst[e] == n) { ++cc; d += ew[e]; }
    cnt[n] = cc; deg[n] = d;
  }
  __syncthreads();
  if (tid == 0) {
    int o = 0;
    for (int n = 0; n < kN; ++n) { off[n] = o; o += cnt[n]; }
    off[kN] = o;
  }
  __syncthreads();
  for (int n = tid; n < kN; n += blockDim.x) {
    int p = off[n];
    for (int e = 0; e < kE; ++e) if (dst[e] == n) cidx[p++] = e;
  }
  for (int e = tid; e < kE; e += blockDim.x) {
    float d = deg[dst[e]];
    nw[e] = d > 0.f ? ew[e] / d : 0.f;
  }
}

__global__ void k_gcn(const float* __restrict__ rootp, const float* __restrict__ xw,
                      const int* __restrict__ edges, const float* __restrict__ nw,
                      const int* __restrict__ off, const int* __restrict__ cidx,
                      float* __restrict__ out)
{
  long long row = blockIdx.x;
  int c = threadIdx.x;
  int n = (int)(row % kN);
  long long btbase = row - n;
  float a = rootp[row * 64 + c];
  const int* src = edges;
  int j0 = off[n], j1 = off[n + 1];
  for (int j = j0; j < j1; ++j) {
    int e = cidx[j];
    a += xw[(btbase + src[e]) * 64 + c] * nw[e];
  }
  out[row * 64 + c] = a;
}

// ---------------------------------------------------------------------------
// Small kernels
// ---------------------------------------------------------------------------
__global__ void k_convin(const float* __restrict__ xa, const float* __restrict__ xc,
                         const float* __restrict__ xm, const float* __restrict__ u,
                         const float* __restrict__ Wx, const float* __restrict__ bx,
                         const float* __restrict__ Wu, const float* __restrict__ bu,
                         float* __restrict__ out)
{
  long long row = blockIdx.x;
  int c = threadIdx.x;
  float x0 = xa[row], x1 = xc[row], x2 = xm[row];
  float v = bx[c] + Wx[c * 3 + 0] * x0 + Wx[c * 3 + 1] * x1 + Wx[c * 3 + 2] * x2;
  if (u) v += bu[c] + Wu[c * 2 + 0] * u[row * 2] + Wu[c * 2 + 1] * u[row * 2 + 1];
  out[row * 64 + c] = v;
}

__global__ void k_tc(const int* __restrict__ t, const float* __restrict__ w1,
                     const float* __restrict__ b1, const float* __restrict__ w2,
                     const float* __restrict__ b2, float* __restrict__ out)
{
  __shared__ float tp[128];
  __shared__ float h1[64];
  int b = blockIdx.x, j = threadIdx.x;
  float tf = (float)t[b];
  if (j < 128) {
    int k = j & 63;
    float f = powf(10000.f, -(float)k / 64.f);
    tp[j] = (j < 64) ? sinf(tf * f) : cosf(tf * f);
  }
  __syncthreads();
  if (j < 64) {
    float s = b1[j];
    for (int k = 0; k < 128; ++k) s += w1[j * 128 + k] * tp[k];
    h1[j] = s / (1.f + expf(-s));
  }
  __syncthreads();
  if (j < 64) {
    float s = b2[j];
    for (int k = 0; k < 64; ++k) s += w2[j * 64 + k] * h1[k];
    out[b * 64 + j] = s / (1.f + expf(-s));
  }
}

__global__ void k_add2(float* __restrict__ y, const float* __restrict__ a,
                       const float* __restrict__ b, int n)
{
  int i = blockIdx.x * blockDim.x + threadIdx.x;
  if (i < n) y[i] = a[i] + b[i];
}

__global__ void k_add3(float* __restrict__ y, const float* __restrict__ a,
                       const float* __restrict__ b, const float* __restrict__ c, int n)
{
  int i = blockIdx.x * blockDim.x + threadIdx.x;
  if (i < n) y[i] = a[i] + b[i] + c[i];
}

__global__ void k_addbc(float* __restrict__ y, const float* __restrict__ a,
                        const float* __restrict__ tc, int n)
{
  int i = blockIdx.x * blockDim.x + threadIdx.x;
  if (i < n) {
    int b = i / (int)kPerB;
    y[i] = a[i] + tc[b * 64 + (i & 63)];
  }
}

__global__ void k_gate(float* __restrict__ y, const float* __restrict__ h128, int n)
{
  int i = blockIdx.x * blockDim.x + threadIdx.x;
  if (i < n) {
    long long r = i >> 6;
    int c = i & 63;
    float g = h128[r * 128 + c];
    float f = h128[r * 128 + 64 + c];
    y[i] = (1.f / (1.f + expf(-g))) * tanhf(f);
  }
}

__global__ void k_residskip(float* __restrict__ hin, float* __restrict__ skip,
                            const float* __restrict__ y128, int n)
{
  int i = blockIdx.x * blockDim.x + threadIdx.x;
  if (i < n) {
    long long r = i >> 6;
    int c = i & 63;
    hin[i]  = (hin[i] + y128[r * 128 + c]) * 0.7071067811865475f;
    skip[i] += y128[r * 128 + 64 + c];
  }
}

__global__ void k_zero(float* __restrict__ y, int n)
{
  int i = blockIdx.x * blockDim.x + threadIdx.x;
  if (i < n) y[i] = 0.f;
}

__global__ void k_out2(const float* __restrict__ x, const float* __restrict__ w,
                       const float* __restrict__ b, float* __restrict__ y, int rows)
{
  int r = blockIdx.x * blockDim.x + threadIdx.x;
  if (r >= rows) return;
  float s = b[0];
  const float* xp = x + (long long)r * 64;
  for (int k = 0; k < 64; ++k) s += xp[k] * w[k];
  y[r] = s;
}

// ---------------------------------------------------------------------------
// Host-side parameter structs (JAX pytree order: dict keys sorted)
// ---------------------------------------------------------------------------
struct MhaP  { const float *Wk, *Wo, *Wq, *Wv, *bk, *bo, *bq, *bv; };
struct TempP { const float *W1, *W2, *b1, *b2, *lng, *lnb, *mlng, *mlnb; MhaP mha; };
struct SpaP  { const float *W1, *W2, *b1, *b2, *Wd, *bd, *We, *be, *Wp, *bp,
                           *Wg, *Wroot, *bg, *gag, *gab, *gfg, *gfb, *glg, *glb; MhaP mha; };
struct NemP  { const float *Wc, *bc, *Wm, *bm, *Wo2, *bo2; SpaP spa;
               const float *t1W, *t1b, *t2W, *t2b; TempP temp; };
struct CfemP { const float *W1, *W2, *b1, *b2, *Wu, *bu, *Wx, *bx, *ng, *nb;
               SpaP spa; TempP temp; };

struct Cur {
  void* const* d;
  int i;
  const float* f() { return (const float*)d[i++]; }
};

void parseMha(Cur& c, MhaP& m) {
  m.Wk = c.f(); m.Wo = c.f(); m.Wq = c.f(); m.Wv = c.f();
  m.bk = c.f(); m.bo = c.f(); m.bq = c.f(); m.bv = c.f();
}
void parseTemp(Cur& c, TempP& t) {
  t.W1 = c.f(); t.W2 = c.f(); t.b1 = c.f(); t.b2 = c.f();
  t.lng = c.f(); t.lnb = c.f(); t.mlng = c.f(); t.mlnb = c.f();
  parseMha(c, t.mha);
}
void parseSpa(Cur& c, SpaP& s) {
  s.W1 = c.f(); s.W2 = c.f(); s.b1 = c.f(); s.b2 = c.f();
  s.Wd = c.f(); s.bd = c.f();
  s.We = c.f(); s.be = c.f();
  s.Wp = c.f(); s.bp = c.f();
  s.Wg = c.f(); s.Wroot = c.f(); s.bg = c.f();
  s.gag = c.f(); s.gab = c.f();
  s.gfg = c.f(); s.gfb = c.f();
  s.glg = c.f(); s.glb = c.f();
  parseMha(c, s.mha);
}

struct Ctx {
  hipStream_t st;
  float* A[11];
  float *W128A, *W128B;
  float *stats, *deg, *nwf, *tc2;
  int *cnt, *offp, *cidx;
  const int* edges;
};

inline int divup(int a, int b) { return (a + b - 1) / b; }

void gemmN(Ctx& c, const float* A, long long aR, long long aC, long long aB,
           const float* Bm, long long bR, long long bC, long long bB,
           const float* bias, int bm, float* C, long long cR, long long cC, long long cB,
           int M, int N, int K, int batch, int act, int acc, float alpha)
{
  int tn = divup(N, 16);
  dim3 grid((unsigned)(divup(M, 16) * tn), (unsigned)batch);
  k_gemm<<<grid, dim3(32), 0, c.st>>>(A, aR, aC, aB, Bm, bR, bC, bB, bias, bm,
                                      C, cR, cC, cB, M, N, K, tn, act, acc, alpha);
}

// Y[M,Nout] = act(alpha * X[M,Kin] @ W[Nout,Kin]^T + bias) (+= if acc)
// Fast path for Nout in {64,128}; M%128==0 and Kin%4==0 always hold here.
void lin(Ctx& c, const float* X, const float* W, const float* bias, float* Y,
         int M, int Nout, int Kin, int act = 0, int acc = 0, float alpha = 1.f)
{
  dim3 grid((unsigned)(M / 128));
  if (Nout == 64) {
    k_lin<4><<<grid, dim3(256), 0, c.st>>>(X, Kin, W, bias, Y, act, acc, alpha);
  } else if (Nout == 128) {
    k_lin<8><<<grid, dim3(256), 0, c.st>>>(X, Kin, W, bias, Y, act, acc, alpha);
  } else {
    gemmN(c, X, Kin, 1, 0, W, 1, Kin, 0, bias, bias ? 1 : 0,
          Y, Nout, 1, 0, M, Nout, Kin, 1, act, acc, alpha);
  }
}

void gnorm(Ctx& c, const float* in1, const float* in2, const float* g, const float* b,
           float* out, int relu)
{
  k_gnstats<<<16, 256, 0, c.st>>>(in1, in2, c.stats);
  int total = kRows * 64;
  k_gnapply<<<divup(total, 256), 256, 0, c.st>>>(in1, in2, c.stats, g, b, out, relu, total);
}

// temp_module: uses A[5..9] internally. out must be distinct from those & v.
void run_temp(Ctx& c, const TempP& t, const float* v, const float* q, const float* k,
              float* out)
{
  int total = kRows * 64, eg = divup(total, 256);
  float *Qp = c.A[5], *Kp = c.A[6], *Vp = c.A[7], *O = c.A[8], *H2 = c.A[9];
  lin(c, q, t.mha.Wq, t.mha.bq, Qp, kRows, 64, 64);
  lin(c, k, t.mha.Wk, t.mha.bk, Kp, kRows, 64, 64);
  lin(c, v, t.mha.Wv, t.mha.bv, Vp, kRows, 64, 64);
  dim3 ag((unsigned)(kB * kN), (unsigned)kH);   // temporal attention: group = (b,n)
  k_attn<<<ag, 64, 0, c.st>>>(Qp, Kp, Vp, O, kN, (long long)kT * kN * 64, 64,
                              (long long)kN * 64);
  lin(c, O, t.mha.Wo, t.mha.bo, Qp, kRows, 64, 64);          // out-proj -> Qp
  k_add2<<<eg, 256, 0, c.st>>>(H2, v, Qp, total);            // h = v + attn
  k_ln<<<divup(kRows, 8), 256, 0, c.st>>>(H2, nullptr, t.mlng, t.mlnb, Kp, kRows); // m
  lin(c, Kp, t.W1, t.b1, Vp, kRows, 64, 64, 2);              // gelu(m@W1+b1)
  lin(c, Vp, t.W2, t.b2, O, kRows, 64, 64);                  // m2
  k_ln<<<divup(kRows, 8), 256, 0, c.st>>>(H2, O, t.lng, t.lnb, out, kRows);
}

// spa_module: uses A[3] (vbuf) and A[5..10]. is_pri -> returns hgcn/hatt ptrs,
// else returns out_final pointer (A[6]).
float* run_spa(Ctx& c, const SpaP& s, const float* h, const float* hpri, bool is_pri,
               float** hgcn_out, float** hatt_out)
{
  float *G0 = c.A[5], *G1 = c.A[6], *G2 = c.A[7], *G3 = c.A[8], *G4 = c.A[9], *G5 = c.A[10];
  float* vbuf = c.A[3];
  const int MR = kB * kT * 64;
  int total = kRows * 64, eg = divup(total, 256);

  // --- graph conv + gn_local
  lin(c, h, s.Wg, nullptr, G0, kRows, 64, 64);               // xw
  lin(c, h, s.Wroot, s.bg, G1, kRows, 64, 64);               // root + bias
  k_gcn<<<kRows, 64, 0, c.st>>>(G1, G0, c.edges, c.nwf, c.offp, c.cidx, G2);
  gnorm(c, G2, h, s.glg, s.glb, G3, 0);                      // h_gcn -> G3

  // --- encoder: v[b,t,s=c,f=k] = sum_n We[f,n]*h[b,t,n,c] + be[f]
  gemmN(c, s.We, 207, 1, 0, h, 64, 1, (long long)kN * 64, s.be, 2,
        vbuf, 1, 64, 4096, 64, 64, 207, kB * kT, 0, 0, 1.f);
  const float* qsrc = vbuf;
  if (!is_pri) {  // q[b,t,i,f] = sum_n Wp[i,n]*h_pri[b,t,n,f] + bp[i]
    gemmN(c, s.Wp, 207, 1, 0, hpri, 64, 1, (long long)kN * 64, s.bp, 2,
          G0, 64, 1, 4096, 64, 64, 207, kB * kT, 0, 0, 1.f);
    qsrc = G0;
  }
  // --- attention over axis 2 (seq=64 per (b,t))
  lin(c, qsrc, s.mha.Wq, s.mha.bq, G1, MR, 64, 64);
  lin(c, qsrc, s.mha.Wk, s.mha.bk, G2, MR, 64, 64);
  lin(c, vbuf, s.mha.Wv, s.mha.bv, G4, MR, 64, 64);
  dim3 ag((unsigned)(kB * kT), (unsigned)kH);
  k_attn<<<ag, 64, 0, c.st>>>(G1, G2, G4, G5, 1, 4096, 0, 64);
  lin(c, G5, s.mha.Wo, s.mha.bo, G1, MR, 64, 64);            // ha -> G1

  // --- decoder: out[b,t,n,k] = sum_x Wd[n,x]*ha[b,t,x,k] + bd[n]
  gemmN(c, s.Wd, 64, 1, 0, G1, 64, 1, 4096, s.bd, 2,
        G2, 64, 1, (long long)kN * 64, kN, 64, 64, kB * kT, 0, 0, 1.f);
  gnorm(c, G2, h, s.gag, s.gab, G4, 0);                      // h_att -> G4

  if (is_pri) { *hgcn_out = G3; *hatt_out = G4; return nullptr; }

  k_add2<<<eg, 256, 0, c.st>>>(G5, G3, G4, total);           // h_comb
  lin(c, G5, s.W1, s.b1, c.W128A, kRows, 128, 64, 1);        // relu mlp
  lin(c, c.W128A, s.W2, s.b2, G0, kRows, 64, 128);
  gnorm(c, G0, G5, s.gfg, s.gfb, G1, 0);                     // gn_final(res + h_comb)
  return G1;
}

}  // namespace

extern "C" void kernel_launch(void* const* d_in, const int* in_sizes, int n_in,
                              void* d_out, int out_size, void* d_ws, size_t ws_size,
                              hipStream_t stream)
{
  (void)in_sizes; (void)n_in; (void)out_size; (void)ws_size;

  const float* x_ta = (const float*)d_in[0];
  const float* x_co = (const float*)d_in[1];
  const float* m_co = (const float*)d_in[2];
  const float* u    = (const float*)d_in[3];
  const int*   tarr = (const int*)d_in[4];
  const int*   edges = (const int*)d_in[5];
  const float* ew    = (const float*)d_in[6];

  Cur cur{d_in, 7};
  CfemP cf;
  cf.W1 = cur.f(); cf.W2 = cur.f(); cf.b1 = cur.f(); cf.b2 = cur.f();
  cf.Wu = cur.f(); cf.bu = cur.f();
  cf.Wx = cur.f(); cf.bx = cur.f();
  cf.ng = cur.f(); cf.nb = cur.f();
  parseSpa(cur, cf.spa);
  parseTemp(cur, cf.temp);
  const float* WiW = cur.f();
  const float* Wib = cur.f();
  NemP nem[4];
  for (int i = 0; i < 4; ++i) {
    NemP& n = nem[i];
    n.Wc = cur.f(); n.bc = cur.f();
    n.Wm = cur.f(); n.bm = cur.f();
    n.Wo2 = cur.f(); n.bo2 = cur.f();
    parseSpa(cur, n.spa);
    n.t1W = cur.f(); n.t1b = cur.f();
    n.t2W = cur.f(); n.t2b = cur.f();
    parseTemp(cur, n.temp);
  }
  const float* o1W = cur.f(); const float* o1b = cur.f();
  const float* o2W = cur.f(); const float* o2b = cur.f();

  Ctx c;
  c.st = stream;
  char* w = (char*)d_ws;
  c.stats = (float*)(w + 0);       // 32 floats
  c.deg   = (float*)(w + 1024);    // 207
  c.nwf   = (float*)(w + 4096);    // 3312
  c.cnt   = (int*)(w + 20480);     // 207
  c.offp  = (int*)(w + 21504);     // 208
  c.cidx  = (int*)(w + 24576);     // 3312
  c.tc2   = (float*)(w + 40960);   // 256
  float* base = (float*)(w + 65536);
  for (int i = 0; i < 11; ++i) c.A[i] = base + (size_t)i * kSlot;
  c.W128A = base + (size_t)11 * kSlot;   // 2 slots wide
  c.W128B = base + (size_t)13 * kSlot;   // 2 slots wide
  c.edges = edges;

  const int TOTAL = kRows * 64;
  const int EG = divup(TOTAL, 256);

  // CSR for graph conv (deterministic)
  k_csr<<<1, 256, 0, stream>>>(edges, ew, c.deg, c.nwf, c.cnt, c.offp, c.cidx);

  // ---------------- CFEM ----------------
  k_convin<<<kRows, 64, 0, stream>>>(x_ta, x_co, m_co, u, cf.Wx, cf.bx, cf.Wu, cf.bu, c.A[3]);
  run_temp(c, cf.temp, c.A[3], c.A[3], c.A[3], c.A[4]);
  k_add2<<<EG, 256, 0, stream>>>(c.A[4], c.A[4], c.A[3], TOTAL);  // h_temp = temp+h
  float *hg = nullptr, *ha = nullptr;
  run_spa(c, cf.spa, c.A[4], nullptr, true, &hg, &ha);
  k_add3<<<EG, 256, 0, stream>>>(c.A[3], c.A[4], hg, ha, TOTAL);  // h_final
  lin(c, c.A[3], cf.W1, cf.b1, c.W128A, kRows, 128, 64, 1);
  lin(c, c.W128A, cf.W2, cf.b2, c.A[5], kRows, 64, 128);
  gnorm(c, c.A[5], c.A[3], cf.ng, cf.nb, c.A[0], 1);              // h_pri = relu(gn)

  // ---------------- input proj + skip init ----------------
  k_convin<<<kRows, 64, 0, stream>>>(x_ta, x_co, m_co, nullptr, WiW, Wib, nullptr, nullptr, c.A[1]);
  k_zero<<<EG, 256, 0, stream>>>(c.A[2], TOTAL);

  // ---------------- NEM layers ----------------
  for (int L = 0; L < 4; ++L) {
    NemP& n = nem[L];
    k_tc<<<kB, 128, 0, stream>>>(tarr, n.t1W, n.t1b, n.t2W, n.t2b, c.tc2);
    k_addbc<<<EG, 256, 0, stream>>>(c.A[3], c.A[1], c.tc2, TOTAL);    // v = h_in + tc
    run_temp(c, n.temp, c.A[3], c.A[0], c.A[0], c.A[4]);              // h_tem
    float* hspa = run_spa(c, n.spa, c.A[4], c.A[0], false, nullptr, nullptr);
    lin(c, hspa, n.Wm, n.bm, c.W128A, kRows, 128, 64);                // mid
    lin(c, c.A[0], n.Wc, n.bc, c.W128A, kRows, 128, 64, 0, 1);        // += cond
    k_gate<<<EG, 256, 0, stream>>>(c.A[5], c.W128A, TOTAL);           // sig*tanh
    lin(c, c.A[5], n.Wo2, n.bo2, c.W128B, kRows, 128, 64);            // out proj
    k_residskip<<<EG, 256, 0, stream>>>(c.A[1], c.A[2], c.W128B, TOTAL);
  }

  // ---------------- output head ----------------
  lin(c, c.A[2], o1W, o1b, c.A[5], kRows, 64, 64, 1, 0, 0.5f);  // relu((skip/2)@W1+b1)
  k_out2<<<divup(kRows, 256), 256, 0, stream>>>(c.A[5], o2W, o2b, (float*)d_out, kRows);
}